// Net_9929964388375
// MI455X (gfx1250) — compile-verified
//
#include <hip/hip_runtime.h>

typedef float v2f __attribute__((ext_vector_type(2)));
typedef float v8f __attribute__((ext_vector_type(8)));

#define BN_EPS 1e-5f

// ---------------------------------------------------------------------------
// Fused 4-GEMM kernel:  H = x@Wl, BG = x@Wf + bf, SK = x@Ws, BGS = x@Wfs
// Block = 64 node rows staged in LDS via GLOBAL_LOAD_ASYNC_TO_LDS_B128
// (ASYNCcnt path). 8 waves sweep the 24 16-column tiles of the four fused
// GEMMs; each wave holds 4 row-tile accumulators so every B (weight)
// fragment load is reused 4x. V_WMMA_F32_16X16X4_F32 keeps fp32-exact
// numerics; the GEMMs (~20 GFLOP total) are far below the matrix roofline.
// ---------------------------------------------------------------------------
template <int FI>
__global__ __launch_bounds__(256)
void film_gemm_kernel(const float* __restrict__ x,
                      const float* __restrict__ Wl, const float* __restrict__ Wf,
                      const float* __restrict__ bf, const float* __restrict__ Ws,
                      const float* __restrict__ Wfs,
                      float* __restrict__ H,  float* __restrict__ BG,
                      float* __restrict__ SK, float* __restrict__ BGS,
                      int n)
{
    extern __shared__ float ldsx[]; // 64 * FI floats
    const int tid  = threadIdx.x;
    const int row0 = blockIdx.x * 64;

    // Async-stage the 64 x FI tile of x into LDS (16B per op per thread).
    // Tail rows clamp to n-1: loads stay in-bounds; those rows are never
    // stored.
    #pragma unroll
    for (int idx = tid * 4; idx < 64 * FI; idx += 256 * 4) {
        int r  = idx / FI;
        int c  = idx - r * FI;          // multiple of 4 -> 16B aligned
        int gr = min(row0 + r, n - 1);
        const float* gp = &x[(size_t)gr * FI + c];
        unsigned lo = (unsigned)(size_t)(&ldsx[idx]);   // LDS byte address
        asm volatile("global_load_async_to_lds_b128 %0, %1, off"
                     :: "v"(lo), "v"(gp) : "memory");
    }
    asm volatile("s_wait_asynccnt 0x0" ::: "memory");
    __syncthreads();

    const int wave = tid >> 5;        // 0..7 (wave32)
    const int lane = tid & 31;
    const int mrow = lane & 15;       // M for A-frag, N for B/C-frag
    const int hi   = lane >> 4;       // 0 or 1
    const int kA   = hi * 2;          // K sub-offset inside a 4-wide step

    // Per-lane LDS A-fragment bases for the 4 row tiles
    const float* A0 = &ldsx[( 0 + mrow) * FI + kA];
    const float* A1 = &ldsx[(16 + mrow) * FI + kA];
    const float* A2 = &ldsx[(32 + mrow) * FI + kA];
    const float* A3 = &ldsx[(48 + mrow) * FI + kA];

    const bool full = (row0 + 64 <= n);   // block-uniform: no store tail

    for (int t = wave; t < 24; t += 8) {   // wave-uniform tile selection
        const float* W; float* D; int width, c0, addb = 0;
        if (t < 4)       { W = Wl;  D = H;   width = 64;  c0 = t * 16; }
        else if (t < 12) { W = Wf;  D = BG;  width = 128; c0 = (t - 4)  * 16; addb = 1; }
        else if (t < 16) { W = Ws;  D = SK;  width = 64;  c0 = (t - 12) * 16; }
        else             { W = Wfs; D = BGS; width = 128; c0 = (t - 16) * 16; }

        // Per-lane weight pointers, stepped by a constant 4*width stride
        const float* Wp = W + (size_t)kA * width + c0 + mrow;
        const float* Wq = Wp + width;
        const size_t wstep = (size_t)4 * width;

        v8f acc0 = {}, acc1 = {}, acc2 = {}, acc3 = {};
        #pragma unroll 4
        for (int k0 = 0; k0 < FI; k0 += 4) {
            v2f b;
            b.x = *Wp;  Wp += wstep;
            b.y = *Wq;  Wq += wstep;
            v2f a0 = *(const v2f*)(A0 + k0);
            v2f a1 = *(const v2f*)(A1 + k0);
            v2f a2 = *(const v2f*)(A2 + k0);
            v2f a3 = *(const v2f*)(A3 + k0);
            acc0 = __builtin_amdgcn_wmma_f32_16x16x4_f32(false, a0, false, b, (short)0, acc0, false, false);
            acc1 = __builtin_amdgcn_wmma_f32_16x16x4_f32(false, a1, false, b, (short)0, acc1, false, false);
            acc2 = __builtin_amdgcn_wmma_f32_16x16x4_f32(false, a2, false, b, (short)0, acc2, false, false);
            acc3 = __builtin_amdgcn_wmma_f32_16x16x4_f32(false, a3, false, b, (short)0, acc3, false, false);
        }
        const float bias = addb ? bf[c0 + mrow] : 0.0f;
        float* Dp = D + (size_t)row0 * width + c0 + mrow;
        if (full) {
            // Common path: straight stores, no exec-mask churn
            #pragma unroll
            for (int rt = 0; rt < 4; ++rt) {
                const v8f acc = (rt == 0) ? acc0 : (rt == 1) ? acc1 : (rt == 2) ? acc2 : acc3;
                #pragma unroll
                for (int r = 0; r < 8; ++r) {
                    int m = rt * 16 + r + hi * 8;
                    Dp[(size_t)m * width] = acc[r] + bias;
                }
            }
        } else {
            #pragma unroll
            for (int rt = 0; rt < 4; ++rt) {
                const v8f acc = (rt == 0) ? acc0 : (rt == 1) ? acc1 : (rt == 2) ? acc2 : acc3;
                #pragma unroll
                for (int r = 0; r < 8; ++r) {
                    int m = rt * 16 + r + hi * 8;
                    if (row0 + m < n) Dp[(size_t)m * width] = acc[r] + bias;
                }
            }
        }
    }
}

// ---------------------------------------------------------------------------
// Edge kernel: msg = act(gamma[dst]*h[src] + beta[dst]); scatter-mean prep.
// 16 threads per edge, 4 channels each (float4 gathers; the 77 MB H/BG
// working set is L2-resident on the 192 MB L2).
// ---------------------------------------------------------------------------
extern "C" __global__ __launch_bounds__(256)
void film_edge_kernel(const int* __restrict__ src, const int* __restrict__ dst,
                      const float* __restrict__ H, const float* __restrict__ BG,
                      float* __restrict__ AGG, float* __restrict__ CNT,
                      int n_edges, int do_relu)
{
    long long gid = (long long)blockIdx.x * blockDim.x + threadIdx.x;
    int e   = (int)(gid >> 4);
    int sub = (int)(gid & 15);
    if (e >= n_edges) return;
    int s = src[e], d = dst[e];
    int c = sub * 4;

    const float4 h4 = *(const float4*)&H [(long long)s * 64  + c];
    const float4 be = *(const float4*)&BG[(long long)d * 128 + c];
    const float4 ga = *(const float4*)&BG[(long long)d * 128 + 64 + c];

    float m0 = ga.x * h4.x + be.x;
    float m1 = ga.y * h4.y + be.y;
    float m2 = ga.z * h4.z + be.z;
    float m3 = ga.w * h4.w + be.w;
    if (do_relu) {
        m0 = fmaxf(m0, 0.0f); m1 = fmaxf(m1, 0.0f);
        m2 = fmaxf(m2, 0.0f); m3 = fmaxf(m3, 0.0f);
    }
    float* ap = &AGG[(long long)d * 64 + c];
    atomicAdd(ap + 0, m0);
    atomicAdd(ap + 1, m1);
    atomicAdd(ap + 2, m2);
    atomicAdd(ap + 3, m3);
    if (sub == 0) atomicAdd(&CNT[d], 1.0f);
}

// ---------------------------------------------------------------------------
// Finalize: y = act(gamma_s*sk + beta_s) + agg/max(cnt,1); fused BN stats
// (per-channel sum / sumsq) via LDS partial reduction + 64 atomics per block.
// ---------------------------------------------------------------------------
extern "C" __global__ __launch_bounds__(256)
void film_finalize_kernel(const float* __restrict__ SK, const float* __restrict__ BGS,
                          const float* __restrict__ AGG, const float* __restrict__ CNT,
                          float* __restrict__ Y, float* __restrict__ BNS,
                          float* __restrict__ BNSQ,
                          int n, int do_relu, int do_bn)
{
    __shared__ float s_sum[4][64];
    __shared__ float s_sq [4][64];
    const int c    = threadIdx.x & 63;
    const int lr   = threadIdx.x >> 6;     // 0..3
    const int base = blockIdx.x * 64;

    float psum = 0.0f, psq = 0.0f;
    for (int r = lr; r < 64; r += 4) {
        int i = base + r;
        if (i < n) {
            float sk      = SK [(long long)i * 64  + c];
            float beta_s  = BGS[(long long)i * 128 + c];
            float gamma_s = BGS[(long long)i * 128 + 64 + c];
            float v = gamma_s * sk + beta_s;
            if (do_relu) v = fmaxf(v, 0.0f);
            float cnt  = CNT[i];
            float mean = AGG[(long long)i * 64 + c] / fmaxf(cnt, 1.0f);
            float y = v + mean;
            Y[(long long)i * 64 + c] = y;
            psum += y;
            psq  += y * y;
        }
    }
    if (do_bn) {
        s_sum[lr][c] = psum;
        s_sq [lr][c] = psq;
        __syncthreads();
        if (lr == 0) {
            float t1 = s_sum[0][c] + s_sum[1][c] + s_sum[2][c] + s_sum[3][c];
            float t2 = s_sq [0][c] + s_sq [1][c] + s_sq [2][c] + s_sq [3][c];
            atomicAdd(&BNS [c], t1);
            atomicAdd(&BNSQ[c], t2);
        }
    }
}

// ---------------------------------------------------------------------------
// BN apply: x' = (y - mu) * rsqrt(var + eps) * g + b
// ---------------------------------------------------------------------------
extern "C" __global__ __launch_bounds__(256)
void film_bn_apply_kernel(const float* __restrict__ Y,
                          const float* __restrict__ BNS, const float* __restrict__ BNSQ,
                          const float* __restrict__ g, const float* __restrict__ b,
                          float* __restrict__ Xn, int n)
{
    long long idx = (long long)blockIdx.x * blockDim.x + threadIdx.x;
    if (idx >= (long long)n * 64) return;
    int c = (int)(idx & 63);
    float inv_n = 1.0f / (float)n;
    float mu  = BNS[c] * inv_n;
    float var = BNSQ[c] * inv_n - mu * mu;
    Xn[idx] = (Y[idx] - mu) * rsqrtf(var + BN_EPS) * g[c] + b[c];
}

// ---------------------------------------------------------------------------
// Orchestration (all on `stream`; hipMemsetAsync is graph-capture safe)
// ---------------------------------------------------------------------------
extern "C" void kernel_launch(void* const* d_in, const int* in_sizes, int n_in,
                              void* d_out, int out_size, void* d_ws, size_t ws_size,
                              hipStream_t stream)
{
    const float* x0  = (const float*)d_in[0];
    const int*   ei  = (const int*)  d_in[1];
    const int    n   = in_sizes[0] / 128;   // 100000
    const int    E   = in_sizes[1] / 2;     // 1200000

    const float* lins  [3] = {(const float*)d_in[2],  (const float*)d_in[3],  (const float*)d_in[4]};
    const float* films [3] = {(const float*)d_in[5],  (const float*)d_in[6],  (const float*)d_in[7]};
    const float* fbias [3] = {(const float*)d_in[8],  (const float*)d_in[9],  (const float*)d_in[10]};
    const float* skips [3] = {(const float*)d_in[11], (const float*)d_in[12], (const float*)d_in[13]};
    const float* fskips[3] = {(const float*)d_in[14], (const float*)d_in[15], (const float*)d_in[16]};
    const float* bng   [2] = {(const float*)d_in[17], (const float*)d_in[18]};
    const float* bnb   [2] = {(const float*)d_in[19], (const float*)d_in[20]};

    float* ws = (float*)d_ws;
    size_t o = 0;
    float* H    = ws + o; o += (size_t)n * 64;
    float* BG   = ws + o; o += (size_t)n * 128;
    float* SK   = ws + o; o += (size_t)n * 64;
    float* BGS  = ws + o; o += (size_t)n * 128;
    float* AGG  = ws + o; o += (size_t)n * 64;
    float* CNT  = ws + o; o += (size_t)n;
    float* Y    = ws + o; o += (size_t)n * 64;
    float* X    = ws + o; o += (size_t)n * 64;
    float* BNS  = ws + o; o += 64;           // BNS and BNSQ contiguous
    float* BNSQ = ws + o; o += 64;

    const int* src = ei;
    const int* dst = ei + E;

    const float* xin = x0;
    for (int l = 0; l < 3; ++l) {
        const int fi = (l == 0) ? 128 : 64;

        hipMemsetAsync(AGG, 0, (size_t)n * 64 * sizeof(float), stream);
        hipMemsetAsync(CNT, 0, (size_t)n * sizeof(float), stream);
        hipMemsetAsync(BNS, 0, 128 * sizeof(float), stream);

        const int    gblocks = (n + 63) / 64;
        const size_t ldsb    = (size_t)64 * fi * sizeof(float);
        if (fi == 128) {
            film_gemm_kernel<128><<<gblocks, 256, ldsb, stream>>>(
                xin, lins[l], films[l], fbias[l], skips[l], fskips[l],
                H, BG, SK, BGS, n);
        } else {
            film_gemm_kernel<64><<<gblocks, 256, ldsb, stream>>>(
                xin, lins[l], films[l], fbias[l], skips[l], fskips[l],
                H, BG, SK, BGS, n);
        }

        const long long ethreads = (long long)E * 16;
        const int eblocks = (int)((ethreads + 255) / 256);
        film_edge_kernel<<<eblocks, 256, 0, stream>>>(
            src, dst, H, BG, AGG, CNT, E, (l < 2) ? 1 : 0);

        const int fblocks = (n + 63) / 64;
        float* yout = (l == 2) ? (float*)d_out : Y;
        film_finalize_kernel<<<fblocks, 256, 0, stream>>>(
            SK, BGS, AGG, CNT, yout, BNS, BNSQ, n,
            (l < 2) ? 1 : 0, (l < 2) ? 1 : 0);

        if (l < 2) {
            const int bblocks = (int)(((long long)n * 64 + 255) / 256);
            film_bn_apply_kernel<<<bblocks, 256, 0, stream>>>(
                Y, BNS, BNSQ, bng[l], bnb[l], X, n);
            xin = X;
        }
    }
}